// GatingAttentionWmlp_5016521802183
// MI455X (gfx1250) — compile-verified
//
#include <hip/hip_runtime.h>
#include <hip/hip_bf16.h>

#define B_    32
#define F_    512
#define H_    8
#define D_    64
#define S_    256
#define TOPK_ 51
#define STILE 64
#define LROW  517   // padded row stride (floats) for dl/attn tile: 517 mod 64 = 5 (coprime) -> conflict-free
#define VROW  66    // padded row stride (floats) for V tile

typedef float v2f __attribute__((ext_vector_type(2)));
typedef float v8f __attribute__((ext_vector_type(8)));

__device__ __forceinline__ float wave_sum_f(float v) {
#pragma unroll
  for (int off = 16; off > 0; off >>= 1) v += __shfl_xor(v, off, 32);
  return v;
}
__device__ __forceinline__ float wave_max_f(float v) {
#pragma unroll
  for (int off = 16; off > 0; off >>= 1) v = fmaxf(v, __shfl_xor(v, off, 32));
  return v;
}
__device__ __forceinline__ int wave_sum_i(int v) {
#pragma unroll
  for (int off = 16; off > 0; off >>= 1) v += __shfl_xor(v, off, 32);
  return v;
}

__global__ void __launch_bounds__(256, 1)
gating_attn_kernel(const float* __restrict__ values,       // [B,F,H,D]
                   const float* __restrict__ data_logits,  // [B,F,H,S]
                   const float* __restrict__ alpha,        // [H,S,F]
                   const float* __restrict__ beta_param,   // [H,S,1]
                   float* __restrict__ out) {              // [B,S,H,D]
  extern __shared__ float lds[];
  float* attnT = lds;                      // [STILE][LROW]  dl tile, then attn tile (in place)
  float* Vs    = lds + STILE * LROW;       // [F_][VROW]

  const int bh   = blockIdx.x;
  const int b    = bh / H_;
  const int h    = bh % H_;
  const int tid  = threadIdx.x;
  const int lane = tid & 31;
  const int wave = tid >> 5;

  // ---- stage V[b,:,h,:] -> LDS (coalesced over d; conflict-free LDS writes) ----
  {
    const int d  = tid & 63;
    const int f0 = tid >> 6;                       // 0..3
    const float* vb = values + ((size_t)b * F_ * H_ + h) * (size_t)D_ + d;
    for (int f = f0; f < F_; f += 4)
      Vs[f * VROW + d] = vb[(size_t)f * (H_ * D_)];
  }

  const float scale = 0.0441941738241592f;  // 1/sqrt(512)

  for (int s0 = 0; s0 < S_; s0 += STILE) {
    __syncthreads();   // attnT reuse across tiles (and first-iter covers nothing harmful)

    // ---- stage dl tile (transposed): dlT[s_local][f] = data_logits[b,f,h,s0+s_local] ----
    {
      const int sl = tid & 63;
      const int f0 = tid >> 6;
      const float* dlp = data_logits + (((size_t)b * F_) * H_ + h) * (size_t)S_ + s0 + sl;
      for (int f = f0; f < F_; f += 4)
        attnT[sl * LROW + f] = dlp[(size_t)f * (H_ * S_)];
    }
    __syncthreads();

    // ---- phase A: per-row normalize + gated logits + exact top-k + softmax (in place) ----
    for (int j = 0; j < 8; ++j) {
      const int r = wave * 8 + j;     // row in tile
      const int s = s0 + r;

      float x[16];
#pragma unroll
      for (int i = 0; i < 16; ++i) x[i] = attnT[r * LROW + lane + 32 * i];

      float sum = 0.f, sq = 0.f;
#pragma unroll
      for (int i = 0; i < 16; ++i) { sum += x[i]; sq += x[i] * x[i]; }
      sum = wave_sum_f(sum);
      sq  = wave_sum_f(sq);
      const float mu      = sum * (1.0f / F_);
      const float var     = sq * (1.0f / F_) - mu * mu;
      const float inv_std = 1.0f / (sqrtf(var + 1e-6f) + 1e-6f);

      const float bp   = beta_param[h * S_ + s];
      const float beta = (bp > 20.0f) ? bp : log1pf(__expf(bp));  // softplus

      const float* ap = alpha + ((size_t)h * S_ + s) * (size_t)F_;
      float    l[16];
      unsigned key[16];
#pragma unroll
      for (int i = 0; i < 16; ++i) {
        l[i] = ap[lane + 32 * i] * scale + beta * (x[i] - mu) * inv_std;
        unsigned u = __float_as_uint(l[i]);
        key[i] = u ^ (unsigned)(((int)u >> 31) | 0x80000000);  // order-preserving map
      }

      // exact 51st-largest via monotone bit search on keys (all lanes converge to same kth)
      unsigned kth = 0u;
#pragma unroll 1
      for (int bit = 31; bit >= 0; --bit) {
        const unsigned cand = kth | (1u << bit);
        int c = 0;
#pragma unroll
        for (int i = 0; i < 16; ++i) c += (key[i] >= cand) ? 1 : 0;
        c = wave_sum_i(c);
        if (c >= TOPK_) kth = cand;
      }

      // masked softmax over kept entries (max element is always kept)
      float m = -3.4e38f;
#pragma unroll
      for (int i = 0; i < 16; ++i) m = fmaxf(m, l[i]);
      m = wave_max_f(m);

      float z = 0.f;
#pragma unroll
      for (int i = 0; i < 16; ++i) {
        const float e = (key[i] >= kth) ? __expf(l[i] - m) : 0.0f;
        l[i] = e;
        z += e;
      }
      z = wave_sum_f(z);
      const float rz = 1.0f / z;
#pragma unroll
      for (int i = 0; i < 16; ++i)
        attnT[r * LROW + lane + 32 * i] = l[i] * rz;
    }
    __syncthreads();

    // ---- phase B: out[s0:s0+64, 0:64] = attn[64,512] @ V[512,64] via fp32 WMMA ----
    {
      const int tn  = wave & 3;        // d-tile 0..3
      const int tm0 = wave >> 2;       // s-tile 0..1 (second tile = tm0+2), shares B fragment
      const int ln   = lane & 15;
      const int koff = (lane >> 4) << 1;     // K sub-offset per WMMA f32 16x16x4 layout
      const int rowA0 = (tm0 * 16 + ln) * LROW;
      const int rowA1 = ((tm0 + 2) * 16 + ln) * LROW;
      const int colB  = tn * 16 + ln;

      v8f c0 = {};
      v8f c1 = {};
      for (int k = 0; k < F_; k += 4) {
        const int kk = k + koff;
        v2f a0, a1, bb;
        a0.x = attnT[rowA0 + kk];
        a0.y = attnT[rowA0 + kk + 1];
        a1.x = attnT[rowA1 + kk];
        a1.y = attnT[rowA1 + kk + 1];
        bb.x = Vs[kk * VROW + colB];
        bb.y = Vs[(kk + 1) * VROW + colB];
        c0 = __builtin_amdgcn_wmma_f32_16x16x4_f32(false, a0, false, bb,
                                                   (short)0, c0, false, false);
        c1 = __builtin_amdgcn_wmma_f32_16x16x4_f32(false, a1, false, bb,
                                                   (short)0, c1, false, false);
      }

      // C/D layout: VGPR r -> M = r (+8 for lanes 16..31), N = lane&15
      const int half = (lane >> 4) << 3;
      const int dcol = tn * 16 + ln;
#pragma unroll
      for (int r = 0; r < 8; ++r) {
        const int sA = s0 + tm0 * 16 + r + half;
        const int sB = s0 + (tm0 + 2) * 16 + r + half;
        out[(((size_t)b * S_ + sA) * H_ + h) * (size_t)D_ + dcol] = c0[r];
        out[(((size_t)b * S_ + sB) * H_ + h) * (size_t)D_ + dcol] = c1[r];
      }
    }
  }
}

extern "C" void kernel_launch(void* const* d_in, const int* in_sizes, int n_in,
                              void* d_out, int out_size, void* d_ws, size_t ws_size,
                              hipStream_t stream) {
  (void)in_sizes; (void)n_in; (void)out_size; (void)d_ws; (void)ws_size;
  const float* values      = (const float*)d_in[0];
  const float* data_logits = (const float*)d_in[1];
  const float* alpha       = (const float*)d_in[2];
  const float* beta_param  = (const float*)d_in[3];
  float* out = (float*)d_out;

  const size_t lds_bytes = (size_t)(STILE * LROW + F_ * VROW) * sizeof(float); // 267,520 B
  hipFuncSetAttribute((const void*)gating_attn_kernel,
                      hipFuncAttributeMaxDynamicSharedMemorySize, (int)lds_bytes);
  gating_attn_kernel<<<B_ * H_, 256, lds_bytes, stream>>>(values, data_logits, alpha,
                                                          beta_param, out);
}